// MultiHeadAttention_68590627717608
// MI455X (gfx1250) — compile-verified
//
#include <hip/hip_runtime.h>
#include <math.h>

// ---------------------------------------------------------------------------
// PLGA multi-head attention for MI455X (gfx1250).
// fp32 V_WMMA_16X16X4 pipeline + double-buffered GLOBAL_LOAD_ASYNC_TO_LDS_B128
// tile staging (ASYNCcnt software pipelining: DMA of tile i+1 overlaps WMMA of
// tile i; in-order async completion makes s_wait_asynccnt <= N exact).
// B=1, S=2048, D=1024, H=16, depth=64, A_dff=1024, NRES=2, slope=0.2
// ---------------------------------------------------------------------------

typedef float v2f __attribute__((ext_vector_type(2)));
typedef float v8f __attribute__((ext_vector_type(8)));

#define SDIM 2048
#define DDIM 1024
#define NH   16
#define DEP  64
#define ADFF 1024
#define PAD  4

// d_out flat offsets (elements), tuple order: out,E,A,pw,attv,ba,avAp,Ep
#define OFF_OUT  0ull
#define OFF_E    2097152ull
#define OFF_A    69206016ull
#define OFF_PW   69271552ull
#define OFF_ATTV 69271616ull
#define OFF_BA   136380480ull
#define OFF_AVAP 136380481ull
#define OFF_EP   136381505ull

__device__ __forceinline__ v8f wmma4(v2f a, v2f b, v8f c) {
  // V_WMMA_F32_16X16X4_F32 : D = A(16x4) * B(4x16) + C(16x16), fp32
  return __builtin_amdgcn_wmma_f32_16x16x4_f32(
      false, a, false, b, (short)0, c, false, false);
}

typedef __attribute__((address_space(3))) float lds_float;

// Async DMA: copy 16 bytes/lane global -> LDS, tracked by ASYNCcnt.
__device__ __forceinline__ void async_copy16(const float* g, float* l) {
  uint32_t loff = (uint32_t)(uintptr_t)(lds_float*)l;
  asm volatile("global_load_async_to_lds_b128 %0, %1, off"
               :: "v"(loff), "v"(g) : "memory");
}
__device__ __forceinline__ void async_wait0()   { asm volatile("s_wait_asynccnt 0" ::: "memory"); }
__device__ __forceinline__ void async_wait_le2(){ asm volatile("s_wait_asynccnt 2" ::: "memory"); }
__device__ __forceinline__ void async_wait_le4(){ asm volatile("s_wait_asynccnt 4" ::: "memory"); }

// ------------------------------------------------------------------ GEMM ---
// Y = X[M,K] @ W[K,N] + bias[N].  Workgroup tile 64x64, 8 waves, each wave
// owns two 16x16 WMMA tiles.  Double-buffered async LDS staging.
// head_major=1 writes Y as [N/64][M][64].
__global__ __launch_bounds__(256)
void gemm_bias_kernel(const float* __restrict__ X, const float* __restrict__ W,
                      const float* __restrict__ bias, float* __restrict__ Y,
                      int M, int N, int K, int head_major) {
  __shared__ __align__(16) float As[2][64][32 + PAD];   // row stride 144B
  __shared__ __align__(16) float Bs[2][32][64 + PAD];   // row stride 272B
  const int tid  = threadIdx.x;
  const int lane = tid & 31, wave = tid >> 5;
  const int half = lane >> 4, lm = lane & 15;
  const int wr = wave >> 1, wc = wave & 1;
  const int mBase = blockIdx.y * 64, nBase = blockIdx.x * 64;

  auto stage = [&](int k0, int buf) {
    for (int c = tid; c < 512; c += 256) {              // A tile: 64x32 floats
      int r = c >> 3, cc = (c & 7) * 4;
      async_copy16(&X[(size_t)(mBase + r) * K + k0 + cc], &As[buf][r][cc]);
    }
    for (int c = tid; c < 512; c += 256) {              // B tile: 32x64 floats
      int r = c >> 4, cc = (c & 15) * 4;
      async_copy16(&W[(size_t)(k0 + r) * N + nBase + cc], &Bs[buf][r][cc]);
    }
  };

  v8f c0 = {}; v8f c1 = {};
  const int nk = K / 32;
  stage(0, 0);                                          // prologue DMA
  for (int i = 0; i < nk; ++i) {
    const int buf = i & 1;
    if (i + 1 < nk) {
      stage((i + 1) * 32, buf ^ 1);                     // prefetch next tile
      async_wait_le4();                                 // tile i resident
    } else {
      async_wait0();
    }
    __syncthreads();
#pragma unroll
    for (int kk = 0; kk < 32; kk += 4) {
      const int ka = kk + 2 * half;
      v2f a;  a.x  = As[buf][wr * 16 + lm][ka];
              a.y  = As[buf][wr * 16 + lm][ka + 1];
      v2f b0; b0.x = Bs[buf][ka][wc * 32 + lm];
              b0.y = Bs[buf][ka + 1][wc * 32 + lm];
      v2f b1; b1.x = Bs[buf][ka][wc * 32 + 16 + lm];
              b1.y = Bs[buf][ka + 1][wc * 32 + 16 + lm];
      c0 = wmma4(a, b0, c0);
      c1 = wmma4(a, b1, c1);
    }
    __syncthreads();                                    // readers done w/ buf
  }
#pragma unroll
  for (int r = 0; r < 8; ++r) {
    int row  = mBase + wr * 16 + r + 8 * half;
    int col0 = nBase + wc * 32 + lm, col1 = col0 + 16;
    float y0 = c0[r] + bias[col0];
    float y1 = c1[r] + bias[col1];
    if (head_major) {
      Y[(size_t)((col0 >> 6) * M + row) * 64 + (col0 & 63)] = y0;
      Y[(size_t)((col1 >> 6) * M + row) * 64 + (col1 & 63)] = y1;
    } else {
      Y[(size_t)row * N + col0] = y0;
      Y[(size_t)row * N + col1] = y1;
    }
  }
}

// ------------------------------------------------- A = Qh^T Qh (per head) --
__global__ __launch_bounds__(256)
void qq_metric_kernel(const float* __restrict__ qh, float* __restrict__ A) {
  __shared__ __align__(16) float Qs[2][32][64 + PAD];   // row stride 272B
  const int tid  = threadIdx.x;
  const int lane = tid & 31, wave = tid >> 5;
  const int half = lane >> 4, lm = lane & 15;
  const int wr = wave >> 1, wc = wave & 1;
  const int h = blockIdx.x;
  const float* Q = qh + (size_t)h * SDIM * DEP;

  auto stage = [&](int s0, int buf) {
    for (int c = tid; c < 512; c += 256) {              // 32x64 floats
      int r = c >> 4, cc = (c & 15) * 4;
      async_copy16(&Q[(size_t)(s0 + r) * DEP + cc], &Qs[buf][r][cc]);
    }
  };

  v8f c0 = {}; v8f c1 = {};
  const int nk = SDIM / 32;
  stage(0, 0);
  for (int i = 0; i < nk; ++i) {
    const int buf = i & 1;
    if (i + 1 < nk) { stage((i + 1) * 32, buf ^ 1); async_wait_le2(); }
    else           { async_wait0(); }
    __syncthreads();
#pragma unroll
    for (int kk = 0; kk < 32; kk += 4) {
      const int ka = kk + 2 * half;
      // A-operand is Q^T: element [d][s] = Q[s][d]
      v2f a;  a.x  = Qs[buf][ka][wr * 16 + lm];
              a.y  = Qs[buf][ka + 1][wr * 16 + lm];
      v2f b0; b0.x = Qs[buf][ka][wc * 32 + lm];
              b0.y = Qs[buf][ka + 1][wc * 32 + lm];
      v2f b1; b1.x = Qs[buf][ka][wc * 32 + 16 + lm];
              b1.y = Qs[buf][ka + 1][wc * 32 + 16 + lm];
      c0 = wmma4(a, b0, c0);
      c1 = wmma4(a, b1, c1);
    }
    __syncthreads();
  }
  float* Ah = A + (size_t)h * DEP * DEP;
#pragma unroll
  for (int r = 0; r < 8; ++r) {
    int row = wr * 16 + r + 8 * half;
    Ah[row * 64 + wc * 32 + lm]      = c0[r];
    Ah[row * 64 + wc * 32 + 16 + lm] = c1[r];
  }
}

// --------------------------- ResLayerA: LN(relu(A@w1+b1)@w2 + b2 + A) ------
__global__ __launch_bounds__(256)
void res_layer_kernel(const float* __restrict__ Ain, float* __restrict__ Aout,
                      const float* __restrict__ w1, const float* __restrict__ b1,
                      const float* __restrict__ w2, const float* __restrict__ b2,
                      const float* __restrict__ g,  const float* __restrict__ beta) {
  __shared__ __align__(16) float Ah [64][64 + PAD];
  __shared__ __align__(16) float W1 [64][32 + PAD];
  __shared__ __align__(16) float W2 [32][64 + PAD];
  __shared__ __align__(16) float Ht [64][32 + PAD];
  __shared__ __align__(16) float ALN[64][64 + PAD];
  const int tid  = threadIdx.x;
  const int lane = tid & 31, wave = tid >> 5;
  const int half = lane >> 4, lm = lane & 15;
  const int wr = wave >> 1, wc = wave & 1;
  const int h = blockIdx.x;
  const float* Ai = Ain + (size_t)h * DEP * DEP;

  for (int c = tid; c < 1024; c += 256) {               // Ah: 64x64 floats
    int r = c >> 4, cc = (c & 15) * 4;
    async_copy16(&Ai[r * 64 + cc], &Ah[r][cc]);
  }

  v8f c0 = {}; v8f c1 = {};
  for (int f0 = 0; f0 < ADFF; f0 += 32) {
    for (int c = tid; c < 512; c += 256) {              // w1[:, f0:f0+32]
      int r = c >> 3, cc = (c & 7) * 4;
      async_copy16(&w1[(size_t)r * ADFF + f0 + cc], &W1[r][cc]);
    }
    for (int c = tid; c < 512; c += 256) {              // w2[f0:f0+32, :]
      int r = c >> 4, cc = (c & 15) * 4;
      async_copy16(&w2[(size_t)(f0 + r) * DEP + cc], &W2[r][cc]);
    }
    async_wait0();
    __syncthreads();
    {   // Ht = relu(Ah @ W1 + b1)   (M=64, N=32, K=64) — one tile per wave
      const int tm = wave >> 1, tn = wave & 1;
      v8f hc = {};
#pragma unroll
      for (int kk = 0; kk < 64; kk += 4) {
        const int ka = kk + 2 * half;
        v2f a; a.x = Ah[tm * 16 + lm][ka]; a.y = Ah[tm * 16 + lm][ka + 1];
        v2f b; b.x = W1[ka][tn * 16 + lm]; b.y = W1[ka + 1][tn * 16 + lm];
        hc = wmma4(a, b, hc);
      }
#pragma unroll
      for (int r = 0; r < 8; ++r) {
        int row = tm * 16 + r + 8 * half, col = tn * 16 + lm;
        float v = hc[r] + b1[f0 + col];
        Ht[row][col] = v > 0.f ? v : 0.f;
      }
    }
    __syncthreads();
#pragma unroll
    for (int kk = 0; kk < 32; kk += 4) {                // c += Ht @ W2  (K=32)
      const int ka = kk + 2 * half;
      v2f a;  a.x  = Ht[wr * 16 + lm][ka];      a.y  = Ht[wr * 16 + lm][ka + 1];
      v2f b0; b0.x = W2[ka][wc * 32 + lm];      b0.y = W2[ka + 1][wc * 32 + lm];
      v2f b1; b1.x = W2[ka][wc * 32 + 16 + lm]; b1.y = W2[ka + 1][wc * 32 + 16 + lm];
      c0 = wmma4(a, b0, c0);
      c1 = wmma4(a, b1, c1);
    }
    __syncthreads();
  }
#pragma unroll
  for (int r = 0; r < 8; ++r) {
    int row = wr * 16 + r + 8 * half;
    int col0 = wc * 32 + lm, col1 = col0 + 16;
    ALN[row][col0] = c0[r] + b2[col0] + Ah[row][col0];
    ALN[row][col1] = c1[r] + b2[col1] + Ah[row][col1];
  }
  __syncthreads();
  float* Ao = Aout + (size_t)h * DEP * DEP;
  if (tid < 64) {                                       // per-row LayerNorm
    float s = 0.f, s2 = 0.f;
    for (int c = 0; c < 64; ++c) { float x = ALN[tid][c]; s += x; s2 += x * x; }
    float m   = s * (1.f / 64.f);
    float var = s2 * (1.f / 64.f) - m * m;
    float inv = rsqrtf(var + 1e-6f);
    for (int c = 0; c < 64; ++c)
      Ao[tid * 64 + c] = (ALN[tid][c] - m) * inv * g[c] + beta[c];
  }
}

// ------------- avAp[h][e] = sum_d a_vec[d]*pow(|A[h][d][e]|+eps, pw[e]) ----
// also copies A, pw, ba, avAp into d_out.
__global__ __launch_bounds__(256)
void powav_kernel(const float* __restrict__ A, const float* __restrict__ pw,
                  const float* __restrict__ a_vec, const float* __restrict__ ba,
                  float* __restrict__ avAp_ws, float* __restrict__ out_base) {
  const int tid = threadIdx.x;
  for (int idx = tid; idx < NH * DEP; idx += 256) {
    int h = idx >> 6, e = idx & 63;
    float p = pw[e], acc = 0.f;
    const float* Ah = A + (size_t)h * DEP * DEP;
    for (int d = 0; d < DEP; ++d)
      acc += a_vec[d] * powf(fabsf(Ah[d * 64 + e]) + 1e-9f, p);
    avAp_ws[idx] = acc;
    out_base[OFF_AVAP + idx] = acc;
  }
  for (int i = tid; i < 64; i += 256) out_base[OFF_PW + i] = pw[i];
  if (tid == 0) out_base[OFF_BA] = ba[0];
  for (int i = tid; i < NH * DEP * DEP; i += 256) out_base[OFF_A + i] = A[i];
}

// ------- E[h][s][t] = sum_d qh[s,d]*avAp[d]*kh[t,d]; Ep = leaky+ba+mask ----
__global__ __launch_bounds__(256)
void logits_kernel(const float* __restrict__ qh, const float* __restrict__ kh,
                   const float* __restrict__ avAp, const float* __restrict__ mask,
                   const float* __restrict__ ba,
                   float* __restrict__ outE, float* __restrict__ outEp) {
  __shared__ __align__(16) float Qs[64][64 + PAD];      // avAp-scaled q rows
  __shared__ __align__(16) float Kt[64][64 + PAD];      // k rows (B-frag transposed)
  const int tid  = threadIdx.x;
  const int lane = tid & 31, wave = tid >> 5;
  const int half = lane >> 4, lm = lane & 15;
  const int wr = wave >> 1, wc = wave & 1;
  const int h = blockIdx.z, sBase = blockIdx.y * 64, tBase = blockIdx.x * 64;
  const float* Q  = qh + (size_t)h * SDIM * DEP;
  const float* Kh = kh + (size_t)h * SDIM * DEP;
  const float* av = avAp + h * DEP;

  for (int c = tid; c < 1024; c += 256) {               // K tile: pure copy
    int r = c >> 4, cc = (c & 15) * 4;
    async_copy16(&Kh[(size_t)(tBase + r) * DEP + cc], &Kt[r][cc]);
  }
  for (int i = tid; i < 4096; i += 256) {               // Q tile: fused scale
    int r = i >> 6, cc = i & 63;
    Qs[r][cc] = Q[(size_t)(sBase + r) * DEP + cc] * av[cc];
  }
  async_wait0();
  __syncthreads();

  v8f c0 = {}; v8f c1 = {};
#pragma unroll
  for (int kk = 0; kk < 64; kk += 4) {
    const int ka = kk + 2 * half;
    v2f a;  a.x  = Qs[wr * 16 + lm][ka];      a.y  = Qs[wr * 16 + lm][ka + 1];
    v2f b0; b0.x = Kt[wc * 32 + lm][ka];      b0.y = Kt[wc * 32 + lm][ka + 1];
    v2f b1; b1.x = Kt[wc * 32 + 16 + lm][ka]; b1.y = Kt[wc * 32 + 16 + lm][ka + 1];
    c0 = wmma4(a, b0, c0);
    c1 = wmma4(a, b1, c1);
  }
  const float bav = ba[0];
#pragma unroll
  for (int r = 0; r < 8; ++r) {
    int srow = sBase + wr * 16 + r + 8 * half;
    int t0 = tBase + wc * 32 + lm, t1 = t0 + 16;
    size_t base = (size_t)h * SDIM * SDIM + (size_t)srow * SDIM;
    float e0 = c0[r], e1 = c1[r];
    outE[base + t0] = e0;
    outE[base + t1] = e1;
    float ep0 = (e0 > 0.f ? e0 : 0.2f * e0) + bav + mask[t0] * (-1e9f);
    float ep1 = (e1 > 0.f ? e1 : 0.2f * e1) + bav + mask[t1] * (-1e9f);
    outEp[base + t0] = ep0;
    outEp[base + t1] = ep1;
  }
}

// ----------------------------- row softmax over Ep -> attv -----------------
__global__ __launch_bounds__(256)
void softmax_kernel(const float* __restrict__ Ep, float* __restrict__ attv) {
  __shared__ float red[256];
  const int tid = threadIdx.x;
  const size_t row = blockIdx.x;
  const float* x = Ep + row * SDIM;
  float*       y = attv + row * SDIM;

  float vals[8];
  float mx = -1e30f;
#pragma unroll
  for (int i = 0; i < 8; ++i) { vals[i] = x[tid + i * 256]; mx = fmaxf(mx, vals[i]); }
  red[tid] = mx; __syncthreads();
  for (int s = 128; s > 0; s >>= 1) {
    if (tid < s) red[tid] = fmaxf(red[tid], red[tid + s]);
    __syncthreads();
  }
  mx = red[0]; __syncthreads();
  float sum = 0.f;
#pragma unroll
  for (int i = 0; i < 8; ++i) { vals[i] = __expf(vals[i] - mx); sum += vals[i]; }
  red[tid] = sum; __syncthreads();
  for (int s = 128; s > 0; s >>= 1) {
    if (tid < s) red[tid] += red[tid + s];
    __syncthreads();
  }
  const float inv = 1.0f / red[0];
#pragma unroll
  for (int i = 0; i < 8; ++i) y[tid + i * 256] = vals[i] * inv;
}

// ---------------- Hnext[s][h*64+d] = sum_t attv[h][s][t] * vh[h][t][d] -----
__global__ __launch_bounds__(256)
void av_kernel(const float* __restrict__ attv, const float* __restrict__ vh,
               float* __restrict__ Hn) {
  __shared__ __align__(16) float Ps[2][64][32 + PAD];
  __shared__ __align__(16) float Vs[2][32][64 + PAD];
  const int tid  = threadIdx.x;
  const int lane = tid & 31, wave = tid >> 5;
  const int half = lane >> 4, lm = lane & 15;
  const int wr = wave >> 1, wc = wave & 1;
  const int h = blockIdx.y, sBase = blockIdx.x * 64;
  const float* P = attv + (size_t)h * SDIM * SDIM;
  const float* V = vh + (size_t)h * SDIM * DEP;

  auto stage = [&](int t0, int buf) {
    for (int c = tid; c < 512; c += 256) {              // P tile 64x32
      int r = c >> 3, cc = (c & 7) * 4;
      async_copy16(&P[(size_t)(sBase + r) * SDIM + t0 + cc], &Ps[buf][r][cc]);
    }
    for (int c = tid; c < 512; c += 256) {              // V tile 32x64
      int r = c >> 4, cc = (c & 15) * 4;
      async_copy16(&V[(size_t)(t0 + r) * DEP + cc], &Vs[buf][r][cc]);
    }
  };

  v8f c0 = {}; v8f c1 = {};
  const int nk = SDIM / 32;
  stage(0, 0);
  for (int i = 0; i < nk; ++i) {
    const int buf = i & 1;
    if (i + 1 < nk) { stage((i + 1) * 32, buf ^ 1); async_wait_le4(); }
    else           { async_wait0(); }
    __syncthreads();
#pragma unroll
    for (int kk = 0; kk < 32; kk += 4) {
      const int ka = kk + 2 * half;
      v2f a;  a.x  = Ps[buf][wr * 16 + lm][ka];
              a.y  = Ps[buf][wr * 16 + lm][ka + 1];
      v2f b0; b0.x = Vs[buf][ka][wc * 32 + lm];
              b0.y = Vs[buf][ka + 1][wc * 32 + lm];
      v2f b1; b1.x = Vs[buf][ka][wc * 32 + 16 + lm];
              b1.y = Vs[buf][ka + 1][wc * 32 + 16 + lm];
      c0 = wmma4(a, b0, c0);
      c1 = wmma4(a, b1, c1);
    }
    __syncthreads();
  }
#pragma unroll
  for (int r = 0; r < 8; ++r) {
    int row = sBase + wr * 16 + r + 8 * half;
    int d0 = wc * 32 + lm, d1 = d0 + 16;
    Hn[(size_t)row * DDIM + h * 64 + d0] = c0[r];
    Hn[(size_t)row * DDIM + h * 64 + d1] = c1[r];
  }
}

// ---------------------------------------------------------------------------
extern "C" void kernel_launch(void* const* d_in, const int* in_sizes, int n_in,
                              void* d_out, int out_size, void* d_ws, size_t ws_size,
                              hipStream_t stream) {
  (void)in_sizes; (void)n_in; (void)out_size; (void)ws_size;

  const float* q      = (const float*)d_in[0];
  const float* k      = (const float*)d_in[1];
  const float* v      = (const float*)d_in[2];
  const float* mask   = (const float*)d_in[3];
  const float* wq_w   = (const float*)d_in[4];
  const float* wq_b   = (const float*)d_in[5];
  const float* wk_w   = (const float*)d_in[6];
  const float* wk_b   = (const float*)d_in[7];
  const float* wv_w   = (const float*)d_in[8];
  const float* wv_b   = (const float*)d_in[9];
  const float* wo_w   = (const float*)d_in[10];
  const float* wo_b   = (const float*)d_in[11];
  const float* rl_w1  = (const float*)d_in[12];  // [2,64,1024]
  const float* rl_b1  = (const float*)d_in[13];  // [2,1024]
  const float* rl_w2  = (const float*)d_in[14];  // [2,1024,64]
  const float* rl_b2  = (const float*)d_in[15];  // [2,64]
  const float* rl_g   = (const float*)d_in[16];  // [2,64]
  const float* rl_bt  = (const float*)d_in[17];  // [2,64]
  const float* pw     = (const float*)d_in[18];
  const float* a_vec  = (const float*)d_in[19];
  const float* ba     = (const float*)d_in[20];

  float* out = (float*)d_out;
  float* ws  = (float*)d_ws;
  float* qh   = ws;                  // 16*2048*64 = 2,097,152
  float* kh   = ws + 2097152;
  float* vh   = ws + 4194304;
  float* Hn   = ws + 6291456;        // [2048,1024]
  float* A0   = ws + 8388608;        // [16,64,64]
  float* A1   = A0 + 65536;
  float* avAp = A1 + 65536;          // [16,64]

  dim3 blk(256);
  dim3 gProj(DDIM / 64, SDIM / 64);  // (16, 32)

  // Q/K/V projections -> head-major [h][s][64]
  gemm_bias_kernel<<<gProj, blk, 0, stream>>>(q, wq_w, wq_b, qh, SDIM, DDIM, DDIM, 1);
  gemm_bias_kernel<<<gProj, blk, 0, stream>>>(k, wk_w, wk_b, kh, SDIM, DDIM, DDIM, 1);
  gemm_bias_kernel<<<gProj, blk, 0, stream>>>(v, wv_w, wv_b, vh, SDIM, DDIM, DDIM, 1);

  // A = Qh^T Qh per head
  qq_metric_kernel<<<NH, blk, 0, stream>>>(qh, A0);

  // two ResLayerA stacks
  res_layer_kernel<<<NH, blk, 0, stream>>>(A0, A1,
      rl_w1 + 0 * DEP * ADFF, rl_b1 + 0 * ADFF,
      rl_w2 + 0 * ADFF * DEP, rl_b2 + 0 * DEP,
      rl_g  + 0 * DEP,        rl_bt + 0 * DEP);
  res_layer_kernel<<<NH, blk, 0, stream>>>(A1, A0,
      rl_w1 + 1 * DEP * ADFF, rl_b1 + 1 * ADFF,
      rl_w2 + 1 * ADFF * DEP, rl_b2 + 1 * DEP,
      rl_g  + 1 * DEP,        rl_bt + 1 * DEP);

  // power-law scaling + copy A/pw/ba/avAp to d_out
  powav_kernel<<<1, blk, 0, stream>>>(A0, pw, a_vec, ba, avAp, out);

  // logits E + Ep (fused LeakyReLU + ba + mask), stored straight to d_out
  logits_kernel<<<dim3(SDIM / 64, SDIM / 64, NH), blk, 0, stream>>>(
      qh, kh, avAp, mask, ba, out + OFF_E, out + OFF_EP);

  // softmax rows: Ep -> attv (both live in d_out)
  softmax_kernel<<<NH * SDIM, blk, 0, stream>>>(out + OFF_EP, out + OFF_ATTV);

  // Hnext = attv @ vh, written in [s][h*64+d] layout for the output GEMM
  av_kernel<<<dim3(SDIM / 64, NH), blk, 0, stream>>>(out + OFF_ATTV, vh, Hn);

  // out = Hnext @ wo + b
  gemm_bias_kernel<<<gProj, blk, 0, stream>>>(Hn, wo_w, wo_b, out + OFF_OUT,
                                              SDIM, DDIM, DDIM, 0);
}